// MinEntropyConsensusLoss_82875688944017
// MI455X (gfx1250) — compile-verified
//
#include <hip/hip_runtime.h>
#include <math.h>
#include <stdint.h>

// ---------------------------------------------------------------------------
// ce[n] = 0.5 * ( LSE(x[n,:]) + LSE(y[n,:]) - max_c (x[n,c]+y[n,c]) )
// out   = mean_n ce[n]
// HBM-bandwidth-bound streaming reduction (512 MiB read, ~23 us floor at
// 23.3 TB/s). Data path: CDNA5 async global->LDS DMA (ASYNCcnt), double
// buffered, non-temporal (stream has zero reuse). No matmul structure ->
// WMMA inapplicable; wave32 shuffles for cross-lane merges.
// ---------------------------------------------------------------------------

#define C_DIM   2048          // columns (fixed by reference)
#define CHUNK   256           // columns per streamed chunk
#define NCHUNK  (C_DIM / CHUNK)
#define WAVES   8             // waves per block == rows per block
#define BLOCK_T (WAVES * 32)

__device__ __forceinline__ uint32_t lds_lo32(const void* p) {
  // low 32 bits of a flat pointer into LDS == wave-window LDS byte address
  return (uint32_t)(uintptr_t)p;
}

// Issue one chunk's DMA: 1 KB of x and 1 KB of y (4 x b128 per wave).
// The inst offset applies to BOTH the LDS and global address, so one LDS
// address VGPR covers both 512B halves.
__device__ __forceinline__ void issue_chunk(uint32_t ldsx, uint32_t ldsy,
                                            uint32_t gofs, const float* xb,
                                            const float* yb) {
  asm volatile(
      "global_load_async_to_lds_b128 %0, %2, %3 th:TH_LOAD_NT\n\t"
      "global_load_async_to_lds_b128 %0, %2, %3 offset:512 th:TH_LOAD_NT\n\t"
      "global_load_async_to_lds_b128 %1, %2, %4 th:TH_LOAD_NT\n\t"
      "global_load_async_to_lds_b128 %1, %2, %4 offset:512 th:TH_LOAD_NT"
      :
      : "v"(ldsx), "v"(ldsy), "v"(gofs), "s"(xb), "s"(yb)
      : "memory");
}

__global__ __launch_bounds__(BLOCK_T) void mecl_rows(
    const float* __restrict__ x, const float* __restrict__ y,
    float* __restrict__ partial) {
  // [wave][buf(2)][tensor(2)][CHUNK] floats = 32 KB
  __shared__ __align__(16) float smem[WAVES * 2 * 2 * CHUNK];
  __shared__ float ce_sh[WAVES];

  const int lane = threadIdx.x & 31;
  const int wave = threadIdx.x >> 5;
  const int row  = blockIdx.x * WAVES + wave;

  float* wbase = &smem[wave * (2 * 2 * CHUNK)];
  // Per-lane LDS byte addresses of the 4 buffers (buf x tensor).
  const uint32_t lbuf[2][2] = {
      {lds_lo32(wbase + 0 * CHUNK) + (uint32_t)lane * 16u,
       lds_lo32(wbase + 1 * CHUNK) + (uint32_t)lane * 16u},
      {lds_lo32(wbase + 2 * CHUNK) + (uint32_t)lane * 16u,
       lds_lo32(wbase + 3 * CHUNK) + (uint32_t)lane * 16u}};

  const uint32_t row_byte = (uint32_t)row * (C_DIM * 4u);  // < 2^28, fits i32
  const uint32_t lane_byte = row_byte + (uint32_t)lane * 16u;

  // ---- prologue: DMA chunk 0 into buffer 0 -------------------------------
  issue_chunk(lbuf[0][0], lbuf[0][1], lane_byte, x, y);

  float mx = -3.402823466e38f, my = -3.402823466e38f, mxy = -3.402823466e38f;
  float sx = 0.f, sy = 0.f;

  #pragma unroll
  for (int c = 0; c < NCHUNK; ++c) {
    const int cb = c & 1;

    if (c + 1 < NCHUNK) {
      // issue next chunk's DMA into the other buffer, then wait until the
      // 4 older (in-order) async ops have landed
      issue_chunk(lbuf[cb ^ 1][0], lbuf[cb ^ 1][1],
                  lane_byte + (uint32_t)(c + 1) * (CHUNK * 4u), x, y);
      asm volatile("s_wait_asynccnt 0x4" ::: "memory");
    } else {
      asm volatile("s_wait_asynccnt 0x0" ::: "memory");
    }

    // each lane reads exactly the 8 x-floats / 8 y-floats its DMA wrote
    const float4* bx = (const float4*)(wbase + (cb * 2 + 0) * CHUNK);
    const float4* by = (const float4*)(wbase + (cb * 2 + 1) * CHUNK);
    float4 x0 = bx[lane], x1 = bx[32 + lane];
    float4 y0 = by[lane], y1 = by[32 + lane];

    // chunk maxes
    float cmx = fmaxf(fmaxf(fmaxf(x0.x, x0.y), fmaxf(x0.z, x0.w)),
                      fmaxf(fmaxf(x1.x, x1.y), fmaxf(x1.z, x1.w)));
    float cmy = fmaxf(fmaxf(fmaxf(y0.x, y0.y), fmaxf(y0.z, y0.w)),
                      fmaxf(fmaxf(y1.x, y1.y), fmaxf(y1.z, y1.w)));
    mxy = fmaxf(mxy,
          fmaxf(fmaxf(fmaxf(x0.x + y0.x, x0.y + y0.y),
                      fmaxf(x0.z + y0.z, x0.w + y0.w)),
                fmaxf(fmaxf(x1.x + y1.x, x1.y + y1.y),
                      fmaxf(x1.z + y1.z, x1.w + y1.w))));
    // chunk exp-sums (1 exp per element)
    float csx = __expf(x0.x - cmx) + __expf(x0.y - cmx) +
                __expf(x0.z - cmx) + __expf(x0.w - cmx) +
                __expf(x1.x - cmx) + __expf(x1.y - cmx) +
                __expf(x1.z - cmx) + __expf(x1.w - cmx);
    float csy = __expf(y0.x - cmy) + __expf(y0.y - cmy) +
                __expf(y0.z - cmy) + __expf(y0.w - cmy) +
                __expf(y1.x - cmy) + __expf(y1.y - cmy) +
                __expf(y1.z - cmy) + __expf(y1.w - cmy);
    // online (m,s) merge: 2 exps per chunk
    float mn = fmaxf(mx, cmx);
    sx = sx * __expf(mx - mn) + csx * __expf(cmx - mn);
    mx = mn;
    mn = fmaxf(my, cmy);
    sy = sy * __expf(my - mn) + csy * __expf(cmy - mn);
    my = mn;
  }

  // wave32 cross-lane merge (5 xor-shuffle steps)
  #pragma unroll
  for (int o = 16; o > 0; o >>= 1) {
    float mo = __shfl_xor(mx, o, 32);
    float so = __shfl_xor(sx, o, 32);
    float mn = fmaxf(mx, mo);
    sx = sx * __expf(mx - mn) + so * __expf(mo - mn);
    mx = mn;
    mo = __shfl_xor(my, o, 32);
    so = __shfl_xor(sy, o, 32);
    mn = fmaxf(my, mo);
    sy = sy * __expf(my - mn) + so * __expf(mo - mn);
    my = mn;
    mxy = fmaxf(mxy, __shfl_xor(mxy, o, 32));
  }

  if (lane == 0) {
    float lsex = mx + __logf(sx);
    float lsey = my + __logf(sy);
    ce_sh[wave] = 0.5f * (lsex + lsey - mxy);
  }
  __syncthreads();
  if (threadIdx.x == 0) {
    float s = 0.f;
    #pragma unroll
    for (int w = 0; w < WAVES; ++w) s += ce_sh[w];
    partial[blockIdx.x] = s;   // per-block partial sum of ce
  }
}

// Deterministic final mean over the 4096 block partials.
__global__ __launch_bounds__(1024) void mecl_final(
    const float* __restrict__ partial, float* __restrict__ out,
    int nparts, float inv_n) {
  __shared__ float red[32];
  float s = 0.f;
  for (int i = threadIdx.x; i < nparts; i += 1024) s += partial[i];
  #pragma unroll
  for (int o = 16; o > 0; o >>= 1) s += __shfl_xor(s, o, 32);
  if ((threadIdx.x & 31) == 0) red[threadIdx.x >> 5] = s;
  __syncthreads();
  if (threadIdx.x < 32) {
    float v = red[threadIdx.x];
    #pragma unroll
    for (int o = 16; o > 0; o >>= 1) v += __shfl_xor(v, o, 32);
    if (threadIdx.x == 0) out[0] = v * inv_n;
  }
}

extern "C" void kernel_launch(void* const* d_in, const int* in_sizes, int n_in,
                              void* d_out, int out_size, void* d_ws,
                              size_t ws_size, hipStream_t stream) {
  const float* x = (const float*)d_in[0];
  const float* y = (const float*)d_in[1];
  float* out = (float*)d_out;
  float* ws  = (float*)d_ws;

  const int n_rows  = in_sizes[0] / C_DIM;   // 32768
  const int nblocks = n_rows / WAVES;        // 4096

  mecl_rows<<<nblocks, BLOCK_T, 0, stream>>>(x, y, ws);
  mecl_final<<<1, 1024, 0, stream>>>(ws, out, nblocks, 1.0f / (float)n_rows);
}